// GraphSAGELayer_69097433858636
// MI455X (gfx1250) — compile-verified
//
#include <hip/hip_runtime.h>
#include <hip/hip_bf16.h>

typedef __attribute__((ext_vector_type(2))) float v2f;
typedef __attribute__((ext_vector_type(8))) float v8f;

#define D 128           // IN_DIM == OUT_DIM
#define LSTR 132        // LDS row stride (floats): bank = (4m+k)%64, conflict-free

// ---------------------------------------------------------------------------
// Kernel 1: zero workspace (neigh_sum + deg, contiguous)
// ---------------------------------------------------------------------------
__global__ __launch_bounds__(256) void sage_zero(float* __restrict__ p, long long n) {
    long long i = (long long)blockIdx.x * 256 + threadIdx.x;
    if (i < n) p[i] = 0.0f;
}

// ---------------------------------------------------------------------------
// Kernel 2: edge scatter-add.  32 threads per edge, float4 per thread.
// neigh_sum is L2-resident (25.6 MB << 192 MB L2) so atomics RMW in L2.
// ---------------------------------------------------------------------------
__global__ __launch_bounds__(256) void sage_scatter(
    const float* __restrict__ h, const int* __restrict__ src,
    const int* __restrict__ dst, float* __restrict__ nsum,
    float* __restrict__ deg, long long total) {
    long long gid = (long long)blockIdx.x * 256 + threadIdx.x;
    if (gid >= total) return;
    int e = (int)(gid >> 5);   // edge id
    int q = (int)(gid & 31);   // feature quad id
    int s = src[e];
    int d = dst[e];
    const float4 v = *(const float4*)(h + (size_t)s * D + q * 4);
    float* np = nsum + (size_t)d * D + q * 4;
    atomicAdd(np + 0, v.x);
    atomicAdd(np + 1, v.y);
    atomicAdd(np + 2, v.z);
    atomicAdd(np + 3, v.w);
    if (q == 0) atomicAdd(deg + d, 1.0f);
}

// ---------------------------------------------------------------------------
// Kernel 3: fused dual GEMM + bias + ReLU using V_WMMA_F32_16X16X4_F32.
// Block = 256 threads (8 waves). Each block: 16 nodes. Wave w owns output
// columns [16w, 16w+16). Two K=128 passes (h@w_self, neigh@w_neigh) into one
// fp32 accumulator -> exact fp32 math on the matrix pipe.
// ---------------------------------------------------------------------------
__global__ __launch_bounds__(256) void sage_gemm(
    const float* __restrict__ h, const float* __restrict__ nsum,
    const float* __restrict__ deg,
    const float* __restrict__ w_self, const float* __restrict__ b_self,
    const float* __restrict__ w_neigh, const float* __restrict__ b_neigh,
    float* __restrict__ out, int N) {
    __shared__ float sh_h[16 * LSTR];
    __shared__ float sh_n[16 * LSTR];

    const int tid  = threadIdx.x;
    const int base = blockIdx.x * 16;

    // Stage 16x128 tiles of h and neigh (= nsum / max(deg,1)) into LDS.
    for (int idx = tid; idx < 16 * D; idx += 256) {
        int r = idx >> 7;
        int c = idx & (D - 1);
        int node = base + r;
        float hv = 0.0f, nv = 0.0f;
        if (node < N) {
            hv = h[(size_t)node * D + c];
            float dg = deg[node];
            dg = dg < 1.0f ? 1.0f : dg;
            nv = nsum[(size_t)node * D + c] / dg;
        }
        sh_h[r * LSTR + c] = hv;
        sh_n[r * LSTR + c] = nv;
    }
    __syncthreads();

    const int wave  = tid >> 5;        // 0..7
    const int lane  = tid & 31;
    const int nbase = wave * 16;       // output column tile
    const int n     = lane & 15;       // N within tile / A row M
    const int half  = lane >> 4;       // 0: K=k0,k0+1   1: K=k0+2,k0+3
    const int koff  = half * 2;

    v8f acc = {};
    const float* sa = sh_h;
    const float* w  = w_self;
    for (int pass = 0; pass < 2; ++pass) {
        for (int k0 = 0; k0 < D; k0 += 4) {
            v2f a, b;
            // A 16x4 fp32 fragment: lane holds row M=n, K pair per half.
            a.x = sa[n * LSTR + k0 + koff];
            a.y = sa[n * LSTR + k0 + koff + 1];
            // B 4x16 fp32 fragment: row K striped across lanes.
            b.x = w[(size_t)(k0 + koff) * D + nbase + n];
            b.y = w[(size_t)(k0 + koff + 1) * D + nbase + n];
            acc = __builtin_amdgcn_wmma_f32_16x16x4_f32(
                /*neg_a=*/false, a, /*neg_b=*/false, b,
                /*c_mod=*/(short)0, acc, /*reuse_a=*/false, /*reuse_b=*/false);
        }
        sa = sh_n;
        w  = w_neigh;
    }

    // Epilogue: bias + ReLU + store per C/D layout (VGPR r: M=r / M=r+8).
    const float bias = b_self[nbase + n] + b_neigh[nbase + n];
    #pragma unroll
    for (int r = 0; r < 8; ++r) {
        int row = base + r + half * 8;
        if (row < N) {
            float v = acc[r] + bias;
            out[(size_t)row * D + nbase + n] = v > 0.0f ? v : 0.0f;
        }
    }
}

// ---------------------------------------------------------------------------
extern "C" void kernel_launch(void* const* d_in, const int* in_sizes, int n_in,
                              void* d_out, int out_size, void* d_ws, size_t ws_size,
                              hipStream_t stream) {
    const float* h       = (const float*)d_in[0];
    const int*   eidx    = (const int*)d_in[1];   // (2, E): row0=src, row1=dst
    const float* w_self  = (const float*)d_in[2];
    const float* b_self  = (const float*)d_in[3];
    const float* w_neigh = (const float*)d_in[4];
    const float* b_neigh = (const float*)d_in[5];
    float*       out     = (float*)d_out;

    const int N = in_sizes[0] / D;
    const int E = in_sizes[1] / 2;

    float* nsum = (float*)d_ws;                  // N*128 floats
    float* deg  = nsum + (size_t)N * D;          // N floats (contiguous)

    // 1) zero neigh_sum + deg
    long long zn = (long long)N * D + N;
    sage_zero<<<(unsigned)((zn + 255) / 256), 256, 0, stream>>>(nsum, zn);

    // 2) scatter-add over edges
    long long st = (long long)E * 32;
    sage_scatter<<<(unsigned)((st + 255) / 256), 256, 0, stream>>>(
        h, eidx, eidx + E, nsum, deg, st);

    // 3) fused dual-GEMM + bias + ReLU
    sage_gemm<<<(N + 15) / 16, 256, 0, stream>>>(
        h, nsum, deg, w_self, b_self, w_neigh, b_neigh, out, N);
}